// MambaBlock_17119739642025
// MI455X (gfx1250) — compile-verified
//
#include <hip/hip_runtime.h>
#include <cstddef>
#include <cstdint>

typedef __attribute__((ext_vector_type(2))) float v2f;
typedef __attribute__((ext_vector_type(8))) float v8f;

#define DMODEL 1024
#define DSTATE 16
#define DINNER 2048
#define DTRANK 64
#define BATCH  2
#define SEQ    1024
#define MTOT   (BATCH * SEQ)   // 2048 token rows

// Async global->LDS copy (CDNA5, ASYNCcnt-tracked); GV addressing mode.
#define ASYNC_LD128(ldsaddr, gaddr)                                   \
  asm volatile("global_load_async_to_lds_b128 %0, %1, off"            \
               :: "v"(ldsaddr), "v"(gaddr) : "memory")
#define WAIT_ASYNC(n) asm volatile("s_wait_asynccnt " #n ::: "memory")

// ---------------------------------------------------------------------------
// LayerNorm: one block per row of 1024
// ---------------------------------------------------------------------------
__global__ __launch_bounds__(256) void layernorm_kernel(
    const float* __restrict__ x, const float* __restrict__ g,
    const float* __restrict__ b, float* __restrict__ o) {
  __shared__ float ssum[256];
  __shared__ float ssq[256];
  const int row = blockIdx.x;
  const int tid = threadIdx.x;
  const float* xr = x + (size_t)row * DMODEL;
  float s = 0.f, q = 0.f;
  for (int c = tid; c < DMODEL; c += 256) {
    float v = xr[c];
    s += v; q += v * v;
  }
  ssum[tid] = s; ssq[tid] = q;
  __syncthreads();
  for (int st = 128; st > 0; st >>= 1) {
    if (tid < st) { ssum[tid] += ssum[tid + st]; ssq[tid] += ssq[tid + st]; }
    __syncthreads();
  }
  const float mean = ssum[0] * (1.f / DMODEL);
  const float var  = ssq[0] * (1.f / DMODEL) - mean * mean;
  const float rstd = rsqrtf(var + 1e-5f);
  for (int c = tid; c < DMODEL; c += 256)
    o[(size_t)row * DMODEL + c] = (xr[c] - mean) * rstd * g[c] + b[c];
}

// ---------------------------------------------------------------------------
// Shared epilogue
// ---------------------------------------------------------------------------
template <int EPI>
__device__ __forceinline__ float apply_epi(float v, float bv) {
  if (EPI == 1) {
    v = 0.5f * v * (1.f + erff(v * 0.70710678118654752f));  // exact GELU
  } else if (EPI == 2) {
    v += bv;
    v = (v > 20.f) ? v : log1pf(__expf(v));                 // softplus
  } else if (EPI == 3) {
    v += bv;
  }
  return v;
}

// ---------------------------------------------------------------------------
// fp32 WMMA GEMM, async double-buffered: C[M,N] = A[M,K] @ W[N,K]^T
// Requires N % 64 == 0, K % 16 == 0, M % 64 == 0.
// block = 128 threads (4 waves), 64x64 tile, each wave a 32x32 sub-tile.
// Tiles flow global -> LDS via GLOBAL_LOAD_ASYNC_TO_LDS_B128 (no VGPR staging);
// stage s+1 copies overlap stage s's 16 WMMAs.
// ---------------------------------------------------------------------------
template <int EPI>
__global__ __launch_bounds__(128) void wmma_gemm_async(
    const float* __restrict__ A, int lda,
    const float* __restrict__ W, int ldw,
    const float* __restrict__ bias,
    float* __restrict__ C, int ldc,
    int N, int K) {
  __shared__ __align__(16) float As[2][64][20];  // stride 20: conflict-free cols
  __shared__ __align__(16) float Bs[2][64][20];  // Bs[.][n][k] = W[n0+n][k0+k]

  const int tid  = threadIdx.x;
  const int lane = tid & 31;
  const int wave = tid >> 5;
  const int half = lane >> 4;
  const int l16  = lane & 15;
  const int m0 = blockIdx.y * 64;
  const int n0 = blockIdx.x * 64;
  const int wm = (wave >> 1) * 32;
  const int wn = (wave & 1) * 32;

  // Each thread owns two 4-float pieces per matrix per stage: (row0,c4),(row0+32,c4)
  const int row0 = tid >> 2;
  const int c4   = (tid & 3) << 2;
  const float* gA0 = A + (size_t)(m0 + row0) * lda + c4;
  const float* gA1 = A + (size_t)(m0 + row0 + 32) * lda + c4;
  const float* gW0 = W + (size_t)(n0 + row0) * ldw + c4;
  const float* gW1 = W + (size_t)(n0 + row0 + 32) * ldw + c4;
  // LDS byte addresses (flat-LDS aperture: offset lives in addr[31:0])
  const unsigned lA0 = (unsigned)(uintptr_t)&As[0][row0][c4];
  const unsigned lA1 = (unsigned)(uintptr_t)&As[0][row0 + 32][c4];
  const unsigned lB0 = (unsigned)(uintptr_t)&Bs[0][row0][c4];
  const unsigned lB1 = (unsigned)(uintptr_t)&Bs[0][row0 + 32][c4];
  const unsigned BUFSTRIDE = 64u * 20u * 4u;  // 5120 B between double buffers

  v8f acc[2][2] = {};

  // Prologue: kick off stage 0 copies.
  ASYNC_LD128(lA0, gA0);
  ASYNC_LD128(lA1, gA1);
  ASYNC_LD128(lB0, gW0);
  ASYNC_LD128(lB1, gW1);

  for (int k0 = 0; k0 < K; k0 += 16) {
    const int cur = (k0 >> 4) & 1;
    if (k0 + 16 < K) {
      // Issue stage s+1 into the other buffer, then drain stage s (in-order
      // completion => "<=4 outstanding" means stage s has landed).
      const unsigned o = (cur ^ 1) ? BUFSTRIDE : 0u;
      const int kn = k0 + 16;
      ASYNC_LD128(lA0 + o, gA0 + kn);
      ASYNC_LD128(lA1 + o, gA1 + kn);
      ASYNC_LD128(lB0 + o, gW0 + kn);
      ASYNC_LD128(lB1 + o, gW1 + kn);
      WAIT_ASYNC(4);
    } else {
      WAIT_ASYNC(0);
    }
    __syncthreads();  // make all waves' async-landed data visible

#pragma unroll
    for (int kk = 0; kk < 16; kk += 4) {
      const int ks = kk + 2 * half;  // lanes 0-15: K=0,1 ; lanes 16-31: K=2,3
      v2f a0 = *(const v2f*)&As[cur][wm + l16][ks];
      v2f a1 = *(const v2f*)&As[cur][wm + 16 + l16][ks];
      v2f b0 = *(const v2f*)&Bs[cur][wn + l16][ks];
      v2f b1 = *(const v2f*)&Bs[cur][wn + 16 + l16][ks];
      acc[0][0] = __builtin_amdgcn_wmma_f32_16x16x4_f32(false, a0, false, b0, (short)0, acc[0][0], false, false);
      acc[0][1] = __builtin_amdgcn_wmma_f32_16x16x4_f32(false, a0, false, b1, (short)0, acc[0][1], false, false);
      acc[1][0] = __builtin_amdgcn_wmma_f32_16x16x4_f32(false, a1, false, b0, (short)0, acc[1][0], false, false);
      acc[1][1] = __builtin_amdgcn_wmma_f32_16x16x4_f32(false, a1, false, b1, (short)0, acc[1][1], false, false);
    }
    __syncthreads();  // buffer `cur` gets overwritten two stages from now
  }

  // store with fused epilogue (C/D layout: vgpr r -> M = r + 8*half)
#pragma unroll
  for (int i = 0; i < 2; ++i) {
#pragma unroll
    for (int j = 0; j < 2; ++j) {
      const int col = n0 + wn + j * 16 + l16;
      const float bv = (EPI == 2 || EPI == 3) ? bias[col] : 0.f;
#pragma unroll
      for (int r = 0; r < 8; ++r) {
        const int rowg = m0 + wm + i * 16 + r + 8 * half;
        C[(size_t)rowg * ldc + col] = apply_epi<EPI>(acc[i][j][r], bv);
      }
    }
  }
}

// ---------------------------------------------------------------------------
// Guarded synchronous WMMA GEMM for N not multiple of 64 (x_proj, N=96).
// ---------------------------------------------------------------------------
template <int EPI>
__global__ __launch_bounds__(128) void wmma_gemm_guarded(
    const float* __restrict__ A, int lda,
    const float* __restrict__ W, int ldw,
    const float* __restrict__ bias,
    float* __restrict__ C, int ldc,
    int N, int K) {
  __shared__ __align__(16) float As[64][20];
  __shared__ __align__(16) float Bs[64][20];

  const int tid  = threadIdx.x;
  const int lane = tid & 31;
  const int wave = tid >> 5;
  const int half = lane >> 4;
  const int l16  = lane & 15;
  const int m0 = blockIdx.y * 64;
  const int n0 = blockIdx.x * 64;
  const int wm = (wave >> 1) * 32;
  const int wn = (wave & 1) * 32;

  v8f acc[2][2] = {};

  for (int k0 = 0; k0 < K; k0 += 16) {
#pragma unroll
    for (int it = 0; it < 2; ++it) {
      const int i   = tid + it * 128;
      const int row = i >> 2;
      const int c4  = (i & 3) << 2;
      float4 av = *(const float4*)(A + (size_t)(m0 + row) * lda + k0 + c4);
      *(float4*)&As[row][c4] = av;
      const int gn = n0 + row;
      float4 wv = make_float4(0.f, 0.f, 0.f, 0.f);
      if (gn < N) wv = *(const float4*)(W + (size_t)gn * ldw + k0 + c4);
      *(float4*)&Bs[row][c4] = wv;
    }
    __syncthreads();
#pragma unroll
    for (int kk = 0; kk < 16; kk += 4) {
      const int ks = kk + 2 * half;
      v2f a0 = *(const v2f*)&As[wm + l16][ks];
      v2f a1 = *(const v2f*)&As[wm + 16 + l16][ks];
      v2f b0 = *(const v2f*)&Bs[wn + l16][ks];
      v2f b1 = *(const v2f*)&Bs[wn + 16 + l16][ks];
      acc[0][0] = __builtin_amdgcn_wmma_f32_16x16x4_f32(false, a0, false, b0, (short)0, acc[0][0], false, false);
      acc[0][1] = __builtin_amdgcn_wmma_f32_16x16x4_f32(false, a0, false, b1, (short)0, acc[0][1], false, false);
      acc[1][0] = __builtin_amdgcn_wmma_f32_16x16x4_f32(false, a1, false, b0, (short)0, acc[1][0], false, false);
      acc[1][1] = __builtin_amdgcn_wmma_f32_16x16x4_f32(false, a1, false, b1, (short)0, acc[1][1], false, false);
    }
    __syncthreads();
  }

#pragma unroll
  for (int i = 0; i < 2; ++i) {
#pragma unroll
    for (int j = 0; j < 2; ++j) {
      const int col = n0 + wn + j * 16 + l16;
      if (col < N) {
        const float bv = (EPI == 2 || EPI == 3) ? bias[col] : 0.f;
#pragma unroll
        for (int r = 0; r < 8; ++r) {
          const int rowg = m0 + wm + i * 16 + r + 8 * half;
          C[(size_t)rowg * ldc + col] = apply_epi<EPI>(acc[i][j][r], bv);
        }
      }
    }
  }
}

// ---------------------------------------------------------------------------
// Depthwise causal conv (width 4) + SiLU.  xp = xz[..., :DINNER]
// ---------------------------------------------------------------------------
__global__ __launch_bounds__(256) void conv_silu_kernel(
    const float* __restrict__ xz, const float* __restrict__ w,
    const float* __restrict__ bias, float* __restrict__ xc) {
  const int idx = blockIdx.x * blockDim.x + threadIdx.x;  // [B, L, DINNER]
  const int d = idx & (DINNER - 1);
  const int l = (idx >> 11) & (SEQ - 1);
  const int b = idx >> 21;
  float acc = bias[d];
  const float* wd = w + d * 4;
#pragma unroll
  for (int t = 0; t < 4; ++t) {
    const int ls = l - 3 + t;
    if (ls >= 0)
      acc += wd[t] * xz[((size_t)(b * SEQ + ls)) * (2 * DINNER) + d];
  }
  xc[idx] = acc / (1.f + __expf(-acc));  // silu
}

// ---------------------------------------------------------------------------
// Selective scan: one thread per (b, d) channel, sequential over L.
// Fuses y = y*silu(z) gating on output.
// ---------------------------------------------------------------------------
__global__ __launch_bounds__(256) void scan_kernel(
    const float* __restrict__ xc, const float* __restrict__ dt,
    const float* __restrict__ x_dbl, const float* __restrict__ xz,
    const float* __restrict__ A_log, const float* __restrict__ Dp,
    float* __restrict__ yg) {
  const int b = blockIdx.x >> 3;                      // 8 blocks per batch
  const int d = ((blockIdx.x & 7) << 8) + threadIdx.x;

  float Ai[DSTATE];
  const float* al = A_log + (size_t)d * DSTATE;
#pragma unroll
  for (int n = 0; n < DSTATE; ++n) Ai[n] = -__expf(al[n]);  // A = -exp(A_log)
  const float Dd = Dp[d];

  float h[DSTATE];
#pragma unroll
  for (int n = 0; n < DSTATE; ++n) h[n] = 0.f;

  for (int l = 0; l < SEQ; ++l) {
    const size_t bl = (size_t)b * SEQ + l;
    const float* row = x_dbl + bl * 96;  // [dt(64) | B(16) | C(16)], block-uniform
    const float u   = xc[bl * DINNER + d];
    const float dtv = dt[bl * DINNER + d];
    const float du  = dtv * u;
    float y = 0.f;
#pragma unroll
    for (int n = 0; n < DSTATE; ++n) {
      const float dA = __expf(dtv * Ai[n]);
      h[n] = dA * h[n] + du * row[DTRANK + n];
      y += h[n] * row[DTRANK + DSTATE + n];
    }
    y += u * Dd;
    const float z = xz[bl * (2 * DINNER) + DINNER + d];
    yg[bl * DINNER + d] = y * (z / (1.f + __expf(-z)));
  }
}

// ---------------------------------------------------------------------------
// Final GLU gate + residual: out = g[:,:1024]*sigmoid(g[:,1024:]) + skip
// ---------------------------------------------------------------------------
__global__ __launch_bounds__(256) void glu_residual_kernel(
    const float* __restrict__ g, const float* __restrict__ x,
    float* __restrict__ out) {
  const int idx = blockIdx.x * blockDim.x + threadIdx.x;  // [MTOT, DMODEL]
  const int c = idx & (DMODEL - 1);
  const int r = idx >> 10;
  const float a  = g[(size_t)r * 2 * DMODEL + c];
  const float gb = g[(size_t)r * 2 * DMODEL + DMODEL + c];
  out[idx] = a * (1.f / (1.f + __expf(-gb))) + x[idx];
}

// ---------------------------------------------------------------------------
extern "C" void kernel_launch(void* const* d_in, const int* in_sizes, int n_in,
                              void* d_out, int out_size, void* d_ws, size_t ws_size,
                              hipStream_t stream) {
  const float* x         = (const float*)d_in[0];
  const float* ln_g      = (const float*)d_in[1];
  const float* ln_b      = (const float*)d_in[2];
  const float* in_proj_w = (const float*)d_in[3];   // (4096, 1024)
  const float* conv_w    = (const float*)d_in[4];   // (2048, 1, 4)
  const float* conv_b    = (const float*)d_in[5];
  const float* x_proj_w  = (const float*)d_in[6];   // (96, 2048)
  const float* dt_proj_w = (const float*)d_in[7];   // (2048, 64)
  const float* dt_proj_b = (const float*)d_in[8];
  const float* A_log     = (const float*)d_in[9];   // (2048, 16)
  const float* Dp        = (const float*)d_in[10];
  const float* out_proj_w= (const float*)d_in[11];  // (1024, 2048)
  const float* glu_w     = (const float*)d_in[12];  // (2048, 1024)
  const float* glu_b     = (const float*)d_in[13];
  float* out = (float*)d_out;

  char* ws = (char*)d_ws;
  const size_t MB = 1024ull * 1024ull;
  float* xn     = (float*)(ws);              //  8 MB  (2048x1024)
  float* xz     = (float*)(ws +  8 * MB);    // 32 MB  (2048x4096)
  float* xc     = (float*)(ws + 40 * MB);    // 16 MB  (2048x2048)
  float* x_dbl  = (float*)(ws + 56 * MB);    // <1 MB  (2048x96)
  float* dt_act = (float*)(ws + 57 * MB);    // 16 MB  (2048x2048)
  float* yg     = (float*)(ws + 73 * MB);    // 16 MB  (2048x2048)
  float* m_act  = xn;   // alias: xn dead after in_proj
  float* g      = xz;   // alias: z consumed by scan before glu writes

  // 1) LayerNorm
  layernorm_kernel<<<MTOT, 256, 0, stream>>>(x, ln_g, ln_b, xn);
  // 2) in_proj: xz = xn @ in_proj_w.T   (M=2048, N=4096, K=1024)
  wmma_gemm_async<0><<<dim3(4096 / 64, MTOT / 64), 128, 0, stream>>>(
      xn, DMODEL, in_proj_w, DMODEL, nullptr, xz, 2 * DINNER, 2 * DINNER, DMODEL);
  // 3) depthwise causal conv + silu
  conv_silu_kernel<<<(BATCH * SEQ * DINNER) / 256, 256, 0, stream>>>(
      xz, conv_w, conv_b, xc);
  // 4) x_proj: x_dbl = xc @ x_proj_w.T   (N=96, K=2048) -- guarded path
  wmma_gemm_guarded<0><<<dim3(2, MTOT / 64), 128, 0, stream>>>(
      xc, DINNER, x_proj_w, DINNER, nullptr, x_dbl, 96, 96, DINNER);
  // 5) dt_proj + bias + softplus   (A = x_dbl[:, :64] with lda=96; N=2048, K=64)
  wmma_gemm_async<2><<<dim3(DINNER / 64, MTOT / 64), 128, 0, stream>>>(
      x_dbl, 96, dt_proj_w, DTRANK, dt_proj_b, dt_act, DINNER, DINNER, DTRANK);
  // 6) selective scan + y*silu(z) gating
  scan_kernel<<<BATCH * (DINNER / 256), 256, 0, stream>>>(
      xc, dt_act, x_dbl, xz, A_log, Dp, yg);
  // 7) out_proj + exact GELU   (N=1024, K=2048)
  wmma_gemm_async<1><<<dim3(DMODEL / 64, MTOT / 64), 128, 0, stream>>>(
      yg, DINNER, out_proj_w, DINNER, nullptr, m_act, DMODEL, DMODEL, DINNER);
  // 8) glu GEMM + bias   (N=2048, K=1024)
  wmma_gemm_async<3><<<dim3(2 * DMODEL / 64, MTOT / 64), 128, 0, stream>>>(
      m_act, DMODEL, glu_w, DMODEL, glu_b, g, 2 * DMODEL, 2 * DMODEL, DMODEL);
  // 9) gate + residual
  glu_residual_kernel<<<(MTOT * DMODEL) / 256, 256, 0, stream>>>(g, x, out);
}